// DTMLayer_40578851012735
// MI455X (gfx1250) — compile-verified
//
#include <hip/hip_runtime.h>
#include <hip/hip_bf16.h>
#include <stdint.h>

#define HW    4096          // 64*64 grid points
#define NOFF  (127 * 127)   // 16129 candidate (di,dj) offsets
#define M0F   0.05f

typedef float v2f __attribute__((ext_vector_type(2)));
typedef float v8f __attribute__((ext_vector_type(8)));

// ---------------------------------------------------------------------------
// Kernel 1: bound[ch] = M0 * sum(x[ch]) via v_wmma_f32_16x16x4_f32.
// One wave per channel. A holds 64 weights/step (16x4 f32 = 2 VGPRs, loaded as
// one coalesced b64 per lane), B = all-ones => D columns hold row sums.
// EXEC is all-1s (full wave, no divergence) as WMMA requires.
// ---------------------------------------------------------------------------
__global__ void dtm_bounds_wmma(const float* __restrict__ x,
                                float* __restrict__ bounds) {
    const int ch   = blockIdx.x;
    const int lane = threadIdx.x;          // 0..31, wave32
    const v2f* p2  = (const v2f*)(x + ch * HW);

    v8f acc = {};
    v2f ones; ones[0] = 1.0f; ones[1] = 1.0f;

    for (int it = 0; it < HW / 64; ++it) {
        v2f a = p2[it * 32 + lane];        // global_load_b64, any bijection ok
        acc = __builtin_amdgcn_wmma_f32_16x16x4_f32(
            /*neg_a=*/false, a, /*neg_b=*/false, ones,
            /*c_mod=*/(short)0, acc, /*reuse_a=*/false, /*reuse_b=*/false);
    }
    // lane L (<16) holds rows 0..7 of column L; lane L+16 holds rows 8..15.
    float s = acc[0] + acc[1] + acc[2] + acc[3] + acc[4] + acc[5] + acc[6] + acc[7];
    s += __shfl_xor(s, 16, 32);            // combine the two row halves
    if (lane == 0) bounds[ch] = M0F * s;
}

// ---------------------------------------------------------------------------
// Kernel 2: shared neighbor ordering by (d2, lin = di*64+dj) -- exactly
// top_k(-dist)'s (distance asc, linear index asc) order. Rank of each offset
// computed in closed form per row ui:
//   #\{uj : ui^2+uj^2 < d2\} = 2*min(m,63)+1 with m = max integer, m^2 < d2-ui^2
// plus tie candidates uj = +/- s when d2-ui^2 = s^2 (tie-break ulin < lin).
// Keys are unique, so ranks form an exact permutation.
// Output per slot: { bits of r2 = (sqrtf(d2))^2 , (di+64)<<8 | (dj+64) }.
// ---------------------------------------------------------------------------
__device__ __forceinline__ int isqrt_exact(int n) {   // n in [0, 7938]
    int s = (int)sqrtf((float)n);
    while (s * s > n) --s;
    while ((s + 1) * (s + 1) <= n) ++s;
    return s;                               // floor(sqrt(n)), exact
}

__global__ void dtm_sort_offsets(uint2* __restrict__ order) {
    int t = blockIdx.x * blockDim.x + threadIdx.x;
    if (t >= NOFF) return;
    int di  = t / 127 - 63;
    int dj  = t % 127 - 63;
    int d2  = di * di + dj * dj;
    int lin = di * 64 + dj;

    int rank = 0;
    for (int ui = -63; ui <= 63; ++ui) {
        int rem = d2 - ui * ui;
        if (rem < 0) continue;
        if (rem == 0) {                      // uj = 0 ties on d2
            rank += (ui * 64 < lin) ? 1 : 0;
            continue;
        }
        int s = isqrt_exact(rem);
        int m = (s * s == rem) ? s - 1 : s;  // largest m with m^2 < rem
        if (m > 63) m = 63;
        rank += 2 * m + 1;                   // uj in [-m, m] strictly closer
        if (s * s == rem && s <= 63) {       // uj = +/- s tie on d2
            rank += (ui * 64 - s < lin) ? 1 : 0;
            rank += (ui * 64 + s < lin) ? 1 : 0;
        }
    }

    float sq = sqrtf((float)d2);             // match reference: sqrt then square
    float r2 = sq * sq;
    uint2 e;
    e.x = __float_as_uint(r2);
    e.y = ((uint32_t)(di + 64) << 8) | (uint32_t)(dj + 64);
    order[rank] = e;
}

// ---------------------------------------------------------------------------
// Kernel 3: main DTM scan. Grid = (channel, point-tile). Each workgroup stages
// its 16KB weight plane into LDS with global_load_async_to_lds_b128 (ASYNCcnt
// path), then each lane walks the uniform distance-sorted offset stream,
// gathering weights from LDS until its bound is crossed:
//   val = S + r2*(bound - W)  ==  cum_dist[k] + r_dist[k]*(bound - cum_w[k]).
// ---------------------------------------------------------------------------
__global__ void dtm_main(const float* __restrict__ x,
                         const uint2* __restrict__ order,
                         const float* __restrict__ bounds,
                         float* __restrict__ out) {
    __shared__ float sw[HW];               // 16 KB weight plane
    const int ch   = blockIdx.x;
    const int tile = blockIdx.y;
    const int t    = threadIdx.x;          // 0..255
    const float* xc = x + ch * HW;

    // Stage 4096 floats = 1024 x b128 async copies (4 per thread).
    for (int pass = 0; pass < 4; ++pass) {
        int idx4 = t + pass * 256;                         // float4 index
        unsigned ldsOff = (unsigned)(uintptr_t)&sw[idx4 * 4];
        const float* gp = xc + idx4 * 4;
        asm volatile("global_load_async_to_lds_b128 %0, %1, off"
                     :: "v"(ldsOff), "v"(gp)
                     : "memory");
    }
    asm volatile("s_wait_asynccnt 0" ::: "memory");
    __syncthreads();

    const float bound = bounds[ch];
    const int p  = tile * 256 + t;
    const int pi = p >> 6;
    const int pj = p & 63;

    float W = 0.0f, S = 0.0f, val = 0.0f;
    bool done = false;
    for (int n = 0; n < NOFF; ++n) {
        uint2 e = order[n];                // uniform across lanes
        if (!done) {
            float r2 = __uint_as_float(e.x);
            int di = (int)(e.y >> 8) - 64;
            int dj = (int)(e.y & 255u) - 64;
            int ni = pi + di, nj = pj + dj;
            if ((unsigned)ni < 64u && (unsigned)nj < 64u) {
                float w  = sw[(ni << 6) + nj];
                float Wn = W + w;
                if (Wn >= bound) {         // cum_w[k] >= bound: stop here
                    val  = S + r2 * (bound - W);
                    done = true;
                } else {
                    W = Wn;
                    S = S + r2 * w;
                }
            }
        }
        if (__all(done)) break;            // per-wave early exit
    }
    if (!done) val = S;                    // unreachable: bound < total weight

    out[ch * HW + p] = sqrtf(val / bound);
}

// ---------------------------------------------------------------------------
extern "C" void kernel_launch(void* const* d_in, const int* in_sizes, int n_in,
                              void* d_out, int out_size, void* d_ws, size_t ws_size,
                              hipStream_t stream) {
    const float* x = (const float*)d_in[0];
    float* out = (float*)d_out;
    const int nch = in_sizes[0] / HW;      // B*C = 12

    // Workspace layout: [0, 16384*8) sorted offsets; then nch bounds.
    uint2* order  = (uint2*)d_ws;
    float* bounds = (float*)((char*)d_ws + (size_t)16384 * sizeof(uint2));

    dtm_bounds_wmma <<<dim3(nch), dim3(32), 0, stream>>>(x, bounds);
    dtm_sort_offsets<<<dim3((NOFF + 255) / 256), dim3(256), 0, stream>>>(order);
    dtm_main        <<<dim3(nch, HW / 256), dim3(256), 0, stream>>>(x, order, bounds, out);
}